// SphericalConvBlock_74191265071498
// MI455X (gfx1250) — compile-verified
//
#include <hip/hip_runtime.h>
#include <math.h>

typedef __attribute__((ext_vector_type(2))) float v2f;
typedef __attribute__((ext_vector_type(8))) float v8f;

#define EPSC     1e-7f
#define TEMP_INV 1.5f      /* 1 / (2/3) */
#define BN_EPS   1e-5f

// ---------------------------------------------------------------------------
// concrete-dropout retain factor: (1 - sigmoid(logit/temp)) / (1 - p)
// ---------------------------------------------------------------------------
__device__ __forceinline__ float cd_factor(float u, float p) {
  float logit = logf(p + EPSC) - logf(1.f - p + EPSC)
              + logf(u + EPSC) - logf(1.f - u + EPSC);
  float drop = 1.f / (1.f + expf(-logit * TEMP_INV));
  return (1.f - drop) / (1.f - p);
}

// ---------------------------------------------------------------------------
// zero fill (float4 vectorized)
// ---------------------------------------------------------------------------
__global__ __launch_bounds__(256) void zero_kernel(float4* __restrict__ p, long n4) {
  long i = (long)blockIdx.x * blockDim.x + threadIdx.x;
  if (i < n4) p[i] = make_float4(0.f, 0.f, 0.f, 0.f);
}

// ---------------------------------------------------------------------------
// h0[b,v,c] = x[b,v,c] * cd_factor(u[b,c], sigmoid(p_logit))
// ---------------------------------------------------------------------------
template <int C, int B>
__global__ __launch_bounds__(256) void drop_scale_kernel(
    const float* __restrict__ x, float* __restrict__ y,
    const float* __restrict__ u, const float* __restrict__ plogit, int V) {
  long i4 = (long)blockIdx.x * blockDim.x + threadIdx.x;
  long total4 = (long)B * V * C / 4;
  if (i4 >= total4) return;
  long flat = i4 * 4;
  int c0 = (int)(flat & (C - 1));
  int b  = (int)(flat / ((long)V * C));
  float p = 1.f / (1.f + expf(-plogit[0]));
  float4 v = ((const float4*)x)[i4];
  float o[4] = {v.x, v.y, v.z, v.w};
#pragma unroll
  for (int j = 0; j < 4; ++j) o[j] *= cd_factor(u[b * C + c0 + j], p);
  ((float4*)y)[i4] = make_float4(o[0], o[1], o[2], o[3]);
}

// ---------------------------------------------------------------------------
// COO scatter SpMM:  dst[b, rows[e], c] += coef * vals[e] * src[b, cols[e], c]
// C, B compile-time -> index math is shifts/masks only. Consecutive threads
// share e -> coalesced row reads / atomics (128B/256B runs).
// ---------------------------------------------------------------------------
template <int C, int B>
__global__ __launch_bounds__(256) void spmm_scatter_kernel(
    const float* __restrict__ src, float* __restrict__ dst,
    const int* __restrict__ rows, const int* __restrict__ cols,
    const float* __restrict__ vals, float coef, int E, int V) {
  long gid = (long)blockIdx.x * blockDim.x + threadIdx.x;
  long total = (long)E * B * C;
  if (gid >= total) return;
  constexpr int BC = B * C;                 // power of two (128 / 256)
  int e = (int)(gid / BC);                  // -> shift
  int r = (int)(gid & (BC - 1));
  int b = r / C;                            // -> shift
  int c = r & (C - 1);
  float w = vals[e] * coef;
  int ro = rows[e];
  int co = cols[e];
  atomicAdd(&dst[((long)b * V + ro) * C + c],
            w * src[((long)b * V + co) * C + c]);
}

// ---------------------------------------------------------------------------
// Chebyshev GEMM on the WMMA fp32 pipe:
//   y[m,n] = ( x0[m,:]·(W0-W2) + x1[m,:]·W1 + x2s[m,:]·W2 )[n] + bias[n]
// with x2s = 2*L*x1 ("-x0" folded into W0).
// LDS holds weights PAIR-INTERLEAVED: wlds[(k/2)*128 + 2n + (k&1)], so a
// lane's B fragment {W[k][n],W[k+1][n]} is one aligned ds_load_b64 into an
// even VGPR pair (no repacking movs). Each wave computes 32(M) x 64(N):
// 8 acc tiles, every B fragment reused for 2 WMMAs. Bias pre-loaded into acc.
// ---------------------------------------------------------------------------
template <int CinT>
__global__ __launch_bounds__(256, 2) void cheb_gemm_wmma(
    const float* __restrict__ x0, const float* __restrict__ x1,
    const float* __restrict__ x2, const float* __restrict__ W,
    const float* __restrict__ bias, float* __restrict__ y, int M) {
  __shared__ float wlds[3 * CinT * 64];     // pair-interleaved Weff
  constexpr int Kdim = 3 * CinT;

  for (int idx = threadIdx.x; idx < Kdim * 64; idx += 256) {
    int k  = idx >> 6;
    int n  = idx & 63;
    int kb = k / CinT;
    int i  = k - kb * CinT;
    float w = W[(kb * CinT + i) * 64 + n];
    if (kb == 0) w -= W[(2 * CinT + i) * 64 + n];    // fold "-x0" of x2
    wlds[(k >> 1) * 128 + 2 * n + (k & 1)] = w;
  }
  __syncthreads();

  const int lane  = threadIdx.x & 31;
  const int nl    = lane & 15;
  const int khalf = (lane >> 4) << 1;       // 0 for lanes 0-15, 2 for 16-31
  const int wave  = threadIdx.x >> 5;
  const int mBase = blockIdx.x * 256 + wave * 32;
  int row0 = mBase + nl;
  int row1 = row0 + 16;
  if (row0 >= M) row0 = M - 1;              // clamp keeps EXEC all-ones
  if (row1 >= M) row1 = M - 1;

  v8f acc[2][4];
#pragma unroll
  for (int t = 0; t < 4; ++t) {
    float bv = bias[t * 16 + nl];
    v8f init;
#pragma unroll
    for (int r = 0; r < 8; ++r) init[r] = bv;
    acc[0][t] = init;
    acc[1][t] = init;
  }

  const float* srcs[3] = {x0, x1, x2};
#pragma unroll
  for (int buf = 0; buf < 3; ++buf) {
    const float* s0 = srcs[buf] + (long)row0 * CinT;
    const float* s1 = srcs[buf] + (long)row1 * CinT;
    const int kpairBase = (buf * CinT) >> 1;
#pragma unroll 2
    for (int kk = 0; kk < CinT; kk += 4) {
      // A fragments: 16x4 f32, lane = (M=nl, K=kk+khalf .. +1), 8B load
      v2f a0 = *(const v2f*)(s0 + kk + khalf);
      v2f a1 = *(const v2f*)(s1 + kk + khalf);
      // B fragments: one ds_load_b64 per tile from interleaved pairs
      const v2f* wp = (const v2f*)&wlds[(kpairBase + ((kk + khalf) >> 1)) * 128];
#pragma unroll
      for (int t = 0; t < 4; ++t) {
        v2f b = wp[t * 16 + nl];
        acc[0][t] = __builtin_amdgcn_wmma_f32_16x16x4_f32(
            false, a0, false, b, (short)0, acc[0][t], false, false);
        acc[1][t] = __builtin_amdgcn_wmma_f32_16x16x4_f32(
            false, a1, false, b, (short)0, acc[1][t], false, false);
      }
    }
  }

  // D layout: vgpr r holds M = r (lanes 0-15) / M = 8+r (lanes 16-31)
  const int rowHi = (lane >> 4) * 8;
#pragma unroll
  for (int h = 0; h < 2; ++h) {
#pragma unroll
    for (int t = 0; t < 4; ++t) {
      int n = t * 16 + nl;
#pragma unroll
      for (int r2 = 0; r2 < 8; ++r2) {
        long m = (long)mBase + h * 16 + rowHi + r2;
        if (m < M) y[m * 64 + n] = acc[h][t][r2];
      }
    }
  }
}

// ---------------------------------------------------------------------------
// BN stats: stats[c] += sum, stats[64+c] += sumsq over all M rows (C=64)
// ---------------------------------------------------------------------------
__global__ __launch_bounds__(256) void bn_stats_kernel(
    const float* __restrict__ y, float* __restrict__ stats, long Mrows) {
  __shared__ float ssum[256], ssq[256];
  int c = threadIdx.x & 63;
  int g = threadIdx.x >> 6;   // 0..3
  float s = 0.f, q = 0.f;
  for (long row = (long)blockIdx.x * 4 + g; row < Mrows; row += (long)gridDim.x * 4) {
    float v = y[row * 64 + c];
    s += v;
    q += v * v;
  }
  ssum[threadIdx.x] = s;
  ssq[threadIdx.x]  = q;
  __syncthreads();
  if (g == 0) {
    s = ssum[c] + ssum[64 + c] + ssum[128 + c] + ssum[192 + c];
    q = ssq[c]  + ssq[64 + c]  + ssq[128 + c]  + ssq[192 + c];
    atomicAdd(&stats[c], s);
    atomicAdd(&stats[64 + c], q);
  }
}

__global__ __launch_bounds__(64) void bn_finalize_kernel(float* __restrict__ stats,
                                                         float invM) {
  int c = threadIdx.x;
  float m = stats[c] * invM;
  float q = stats[64 + c] * invM;
  stats[128 + c] = m;
  stats[192 + c] = rsqrtf(q - m * m + BN_EPS);
}

// ---------------------------------------------------------------------------
// fused BN + ReLU (+ optional concrete-dropout scale for next layer), C=64
// ---------------------------------------------------------------------------
__global__ __launch_bounds__(256) void bn_relu_drop_kernel(
    const float* __restrict__ src, float* __restrict__ dst,
    const float* __restrict__ stats,
    const float* __restrict__ gamma, const float* __restrict__ beta,
    const float* __restrict__ u, const float* __restrict__ plogit,
    int V, int B, int withDrop) {
  long i4 = (long)blockIdx.x * blockDim.x + threadIdx.x;
  long total4 = (long)B * V * 16;            // B*V*64/4
  if (i4 >= total4) return;
  long flat = i4 * 4;
  int c0 = (int)(flat & 63);
  int b  = (int)(flat / ((long)V * 64));
  float4 v = ((const float4*)src)[i4];
  float o[4] = {v.x, v.y, v.z, v.w};
  float p = withDrop ? (1.f / (1.f + expf(-plogit[0]))) : 0.f;
#pragma unroll
  for (int j = 0; j < 4; ++j) {
    int c = c0 + j;
    float xn = (o[j] - stats[128 + c]) * stats[192 + c];
    float r  = fmaxf(xn * gamma[c] + beta[c], 0.f);
    if (withDrop) r *= cd_factor(u[b * 64 + c], p);
    o[j] = r;
  }
  ((float4*)dst)[i4] = make_float4(o[0], o[1], o[2], o[3]);
}

// ---------------------------------------------------------------------------
// launcher
// ---------------------------------------------------------------------------
extern "C" void kernel_launch(void* const* d_in, const int* in_sizes, int n_in,
                              void* d_out, int out_size, void* d_ws, size_t ws_size,
                              hipStream_t stream) {
  const float* x    = (const float*)d_in[0];
  const int*   rows = (const int*)  d_in[1];
  const int*   cols = (const int*)  d_in[2];
  const float* vals = (const float*)d_in[3];
  const float* u1   = (const float*)d_in[4];
  const float* u2   = (const float*)d_in[5];
  const float* p1   = (const float*)d_in[6];
  const float* p2   = (const float*)d_in[7];
  const float* W1   = (const float*)d_in[8];
  const float* b1   = (const float*)d_in[9];
  const float* W2   = (const float*)d_in[10];
  const float* b2   = (const float*)d_in[11];
  const float* g1   = (const float*)d_in[12];
  const float* be1  = (const float*)d_in[13];
  const float* g2   = (const float*)d_in[14];
  const float* be2  = (const float*)d_in[15];

  const int  B    = 4;
  const int  Cin  = in_sizes[4] / B;            // 32
  const int  Cmid = in_sizes[5] / B;            // 64
  const int  V    = in_sizes[0] / (B * Cin);    // 49152
  const int  E    = in_sizes[1];                // 442368
  const long M    = (long)B * V;                // 196608 (multiple of 256)

  float* ws = (float*)d_ws;
  const long SZ32 = M * Cin;
  const long SZ64 = M * Cmid;
  float* H0    = ws;
  float* X1A   = ws + SZ32;
  float* X2A   = ws + 2 * SZ32;
  float* Y1    = ws + 3 * SZ32;
  float* X2B   = ws + 3 * SZ32 + SZ64;
  float* STATS = ws + 3 * SZ32 + 2 * SZ64;      // 256 floats
  float* H1    = ws;                            // aliases H0/X1A/X2A (dead)
  float* X1B   = Y1;                            // aliases Y1 (dead)
  float* OUT   = (float*)d_out;

  auto cdiv = [](long a, long b) { return (unsigned)((a + b - 1) / b); };

  // ---- layer 1 ----
  zero_kernel<<<cdiv(2 * SZ32 / 4, 256), 256, 0, stream>>>((float4*)X1A, 2 * SZ32 / 4);
  drop_scale_kernel<32, 4><<<cdiv(SZ32 / 4, 256), 256, 0, stream>>>(x, H0, u1, p1, V);

  long tot1 = (long)E * B * Cin;
  spmm_scatter_kernel<32, 4><<<cdiv(tot1, 256), 256, 0, stream>>>(H0, X1A, rows, cols, vals, 1.f, E, V);
  spmm_scatter_kernel<32, 4><<<cdiv(tot1, 256), 256, 0, stream>>>(X1A, X2A, rows, cols, vals, 2.f, E, V);

  cheb_gemm_wmma<32><<<cdiv(M, 256), 256, 0, stream>>>(H0, X1A, X2A, W1, b1, Y1, (int)M);

  zero_kernel<<<1, 64, 0, stream>>>((float4*)STATS, 64);
  bn_stats_kernel<<<1024, 256, 0, stream>>>(Y1, STATS, M);
  bn_finalize_kernel<<<1, 64, 0, stream>>>(STATS, 1.f / (float)M);
  bn_relu_drop_kernel<<<cdiv(SZ64 / 4, 256), 256, 0, stream>>>(Y1, H1, STATS, g1, be1, u2, p2, V, B, 1);

  // ---- layer 2 ----
  zero_kernel<<<cdiv(2 * SZ64 / 4, 256), 256, 0, stream>>>((float4*)X1B, 2 * SZ64 / 4);

  long tot2 = (long)E * B * Cmid;
  spmm_scatter_kernel<64, 4><<<cdiv(tot2, 256), 256, 0, stream>>>(H1, X1B, rows, cols, vals, 1.f, E, V);
  spmm_scatter_kernel<64, 4><<<cdiv(tot2, 256), 256, 0, stream>>>(X1B, X2B, rows, cols, vals, 2.f, E, V);

  cheb_gemm_wmma<64><<<cdiv(M, 256), 256, 0, stream>>>(H1, X1B, X2B, W2, b2, OUT, (int)M);

  zero_kernel<<<1, 64, 0, stream>>>((float4*)STATS, 64);
  bn_stats_kernel<<<1024, 256, 0, stream>>>(OUT, STATS, M);
  bn_finalize_kernel<<<1, 64, 0, stream>>>(STATS, 1.f / (float)M);
  bn_relu_drop_kernel<<<cdiv(SZ64 / 4, 256), 256, 0, stream>>>(OUT, OUT, STATS, g2, be2, nullptr, nullptr, V, B, 0);
}